// MesoNet_40879498729137
// MI455X (gfx1250) — compile-verified
//
#include <hip/hip_runtime.h>
#include <math.h>

// ---------------------------------------------------------------------------
// MesoNet forward for MI455X (gfx1250, wave32).
// All dense contractions (incl. fused NNConv) -> fp32 WMMA
// (V_WMMA_F32_16X16X4_F32). Segment ops via atomics + wave32 shuffles.
// ---------------------------------------------------------------------------

#define ACT_NONE 0
#define ACT_RELU 1
#define ACT_LEAKY 2
#define ACT_ELU 3

typedef __attribute__((ext_vector_type(2))) float v2f;
typedef __attribute__((ext_vector_type(8))) float v8f;

__device__ __forceinline__ v8f wmma_f32(v2f a, v2f b, v8f c) {
#if __has_builtin(__builtin_amdgcn_wmma_f32_16x16x4_f32)
  return __builtin_amdgcn_wmma_f32_16x16x4_f32(false, a, false, b, (short)0, c,
                                               false, false);
#else
  return c;  // never taken on gfx1250 toolchains (probe-verified builtin)
#endif
}

__device__ __forceinline__ float actf(float v, int act) {
  switch (act) {
    case ACT_RELU:  return v > 0.f ? v : 0.f;
    case ACT_LEAKY: return v > 0.f ? v : 0.01f * v;
    case ACT_ELU:   return v > 0.f ? v : (expf(v) - 1.f);
    default:        return v;
  }
}

__device__ __forceinline__ float waveSum(float v) {
  for (int o = 16; o > 0; o >>= 1) v += __shfl_xor(v, o, 32);
  return v;
}
__device__ __forceinline__ float waveMax(float v) {
  for (int o = 16; o > 0; o >>= 1) v = fmaxf(v, __shfl_xor(v, o, 32));
  return v;
}

__device__ __forceinline__ void atomicMaxF(float* addr, float val) {
  int* ia = (int*)addr;
  int old = *ia;
  while (__int_as_float(old) < val) {
    int assumed = old;
    old = atomicCAS(ia, assumed, __float_as_int(val));
    if (old == assumed) break;
  }
}

// ---------------------------------------------------------------------------
// Generic fp32 WMMA GEMM:  C[M,N] = act(A[M,K] @ B[K,N] + bias + addend)
// One wave32 per 16x16 C tile; K marched in steps of 4.
// Fast path (interior tile, K%4==0): zero bounds checks -> clean
// load/load/load/load/wmma inner loop. Guarded path only on fringe tiles.
// ---------------------------------------------------------------------------
__global__ void k_gemm(const float* __restrict__ A, int lda,
                       const float* __restrict__ B, int ldb,
                       const float* __restrict__ bias,
                       const float* __restrict__ addend, int ldadd,
                       float* __restrict__ C, int ldc,
                       int M, int K, int N, int act) {
  const int lane  = threadIdx.x;
  const int m0    = blockIdx.y * 16;
  const int n0    = blockIdx.x * 16;
  const int mrow  = m0 + (lane & 15);
  const int ncol  = n0 + (lane & 15);
  const int khalf = lane >> 4;  // 0 or 1

  v8f acc = {};
  const bool interior = (m0 + 16 <= M) && (n0 + 16 <= N) && ((K & 3) == 0);
  if (interior) {
    const float* Ap = A + mrow * lda + khalf * 2;
    const float* Bp = B + (khalf * 2) * ldb + ncol;
    const int ldb4 = 4 * ldb;
    for (int k0 = 0; k0 < K; k0 += 4) {
      v2f a, b;
      a.x = Ap[0];
      a.y = Ap[1];
      b.x = Bp[0];
      b.y = Bp[ldb];
      acc = wmma_f32(a, b, acc);
      Ap += 4;
      Bp += ldb4;
    }
  } else {
    for (int k0 = 0; k0 < K; k0 += 4) {
      const int ka0 = k0 + khalf * 2;
      const int ka1 = ka0 + 1;
      v2f a, b;
      a.x = (mrow < M && ka0 < K) ? A[mrow * lda + ka0] : 0.f;
      a.y = (mrow < M && ka1 < K) ? A[mrow * lda + ka1] : 0.f;
      b.x = (ka0 < K && ncol < N) ? B[ka0 * ldb + ncol] : 0.f;
      b.y = (ka1 < K && ncol < N) ? B[ka1 * ldb + ncol] : 0.f;
      acc = wmma_f32(a, b, acc);
    }
  }

  const float bb = (bias != nullptr && ncol < N) ? bias[ncol] : 0.f;
  for (int r = 0; r < 8; ++r) {
    const int m = m0 + r + khalf * 8;
    if (m < M && ncol < N) {
      float v = acc[r] + bb;
      if (addend != nullptr) v += addend[m * ldadd + ncol];
      C[m * ldc + ncol] = actf(v, act);
    }
  }
}

// ---------------------------------------------------------------------------
// Fused NNConv on WMMA.  msg[e,o] = sum_k h[e,k] sum_i x[e,i] ew2[k,i,o]
//                                 = (KhatriRao(h,x) @ ew2)[e,o]
// where ew2's native (32, cin*cout) row-major buffer IS the (32*cin, cout)
// row-major B matrix. A-fragments (Atilde[e, k*cin+i] = h[e,k]*x[e,i]) are
// synthesized on the fly from LDS tiles -> 123MB operand never materialized,
// per-edge weight never touches memory. 16 edges per block, one wave per
// 16-wide cout tile (cout/16 waves per block). eb2 bias term folded in as a
// second short WMMA loop over (cin x cout).
// ---------------------------------------------------------------------------
__global__ void k_nnconv_wmma(const float* __restrict__ x, int ldx,
                              const int* __restrict__ src, const int* __restrict__ dst,
                              const float* __restrict__ ea,
                              const float* __restrict__ ew1, const float* __restrict__ eb1,
                              const float* __restrict__ ew2, const float* __restrict__ eb2,
                              float* __restrict__ msum, int cin, int cout, int E) {
  __shared__ float shx[16 * 160];
  __shared__ float shh[16 * 32];
  __shared__ int   shd[16];
  const int e0   = blockIdx.x * 16;
  const int tid  = threadIdx.x;
  const int lane = tid & 31;
  const int wave = tid >> 5;

  // stage x rows for 16 edges
  for (int idx = tid; idx < 16 * cin; idx += blockDim.x) {
    const int er = idx / cin, i = idx - er * cin;
    const int e = e0 + er;
    shx[er * cin + i] = (e < E) ? x[src[e] * ldx + i] : 0.f;
  }
  // h = relu(ea @ ew1 + eb1), 16 edges x 32
  for (int idx = tid; idx < 16 * 32; idx += blockDim.x) {
    const int er = idx >> 5, j = idx & 31;
    const int e = e0 + er;
    float acc = 0.f;
    if (e < E) {
      acc = eb1[j];
      for (int k = 0; k < 10; ++k) acc += ea[e * 10 + k] * ew1[k * 32 + j];
      acc = acc > 0.f ? acc : 0.f;
    }
    shh[idx] = acc;
  }
  if (tid < 16) shd[tid] = (e0 + tid < E) ? dst[e0 + tid] : -1;
  __syncthreads();

  const int n0    = wave * 16;          // this wave's cout tile
  const int ncol  = n0 + (lane & 15);
  const int erow  = lane & 15;          // A-fragment row (edge within tile)
  const int khalf = lane >> 4;
  const float* xrow = &shx[erow * cin];
  const float* hrow = &shh[erow * 32];

  v8f acc = {};
  // main contraction over KK = 32*cin (always a multiple of 4)
  const int KK = 32 * cin;
  int k0 = 0, i0 = khalf * 2;           // decomposition of kka = k0*cin + i0
  const float* Bp = ew2 + (khalf * 2) * cout + ncol;
  const int cout4 = 4 * cout;
  for (int kk0 = 0; kk0 < KK; kk0 += 4) {
    int k1, i1;
    if (i0 + 1 == cin) { k1 = k0 + 1; i1 = 0; }
    else               { k1 = k0;     i1 = i0 + 1; }
    v2f a, b;
    a.x = hrow[k0] * xrow[i0];
    a.y = hrow[k1] * xrow[i1];
    b.x = Bp[0];
    b.y = Bp[cout];
    acc = wmma_f32(a, b, acc);
    Bp += cout4;
    i0 += 4;
    if (i0 >= cin) { i0 -= cin; k0 += 1; }
  }
  // bias term: msg += X @ eb2  (eb2 viewed as cin x cout)
  for (int ib0 = 0; ib0 < cin; ib0 += 4) {
    const int ia = ib0 + khalf * 2, ib = ia + 1;
    v2f a, b;
    a.x = (ia < cin) ? xrow[ia] : 0.f;
    a.y = (ib < cin) ? xrow[ib] : 0.f;
    b.x = (ia < cin) ? eb2[ia * cout + ncol] : 0.f;
    b.y = (ib < cin) ? eb2[ib * cout + ncol] : 0.f;
    acc = wmma_f32(a, b, acc);
  }
  // scatter into segment-sum accumulator
  for (int r = 0; r < 8; ++r) {
    const int d = shd[r + khalf * 8];
    if (d >= 0) atomicAdd(&msum[d * cout + ncol], acc[r]);
  }
}

__global__ void k_nnconv_fin(const float* __restrict__ msum, const float* __restrict__ cnt,
                             const float* __restrict__ rootT, const float* __restrict__ bias,
                             float* __restrict__ out, int Nn, int cout) {
  int idx = blockIdx.x * blockDim.x + threadIdx.x;
  if (idx >= Nn * cout) return;
  const int n = idx / cout, o = idx - n * cout;
  const float v = msum[idx] / fmaxf(cnt[n], 1.f) + rootT[idx] + bias[o];
  out[idx] = v > 0.f ? v : 0.f;
}

// ---------------------------------------------------------------------------
__global__ void k_fill(float* __restrict__ p, float v, int n) {
  int i = blockIdx.x * blockDim.x + threadIdx.x;
  if (i < n) p[i] = v;
}

__global__ void k_count_seg(const int* __restrict__ seg, float* __restrict__ cnt, int n) {
  int i = blockIdx.x * blockDim.x + threadIdx.x;
  if (i < n) atomicAdd(&cnt[seg[i]], 1.f);
}

// CfC recurrence: per-node thread, 5 iterations with 18->12 matrices.
__global__ void k_cfc(const float* __restrict__ x2, int ldx,
                      const float* __restrict__ wf1, const float* __restrict__ bf1,
                      const float* __restrict__ wf2, const float* __restrict__ bf2,
                      const float* __restrict__ wta, const float* __restrict__ bta,
                      const float* __restrict__ wtb, const float* __restrict__ btb,
                      float* __restrict__ out30, int Nn) {
  int n = blockIdx.x * blockDim.x + threadIdx.x;
  if (n >= Nn) return;
  float xv[6], h[12], z[18], hn[12];
  for (int i = 0; i < 6; ++i) xv[i] = x2[n * ldx + i];
  for (int i = 0; i < 6; ++i) { h[i] = xv[i]; h[6 + i] = xv[i]; }
  for (int it = 0; it < 5; ++it) {
    for (int i = 0; i < 6; ++i)  z[i] = xv[i];
    for (int i = 0; i < 12; ++i) z[6 + i] = h[i];
    for (int j = 0; j < 12; ++j) {
      float s1 = bf1[j], s2 = bf2[j], st = bta[j] + btb[j];
      for (int i = 0; i < 18; ++i) {
        const float zi = z[i];
        s1 += zi * wf1[i * 12 + j];
        s2 += zi * wf2[i * 12 + j];
        st += zi * (wta[i * 12 + j] + wtb[i * 12 + j]);
      }
      const float ti = 1.f / (1.f + expf(-st));
      const float l1 = 1.7159f * tanhf(0.666f * s1);
      const float l2 = 1.7159f * tanhf(0.666f * s2);
      hn[j] = l1 * (1.f - ti) + ti * l2;
    }
    for (int j = 0; j < 12; ++j) h[j] = hn[j];
    for (int j = 0; j < 6; ++j) out30[n * 30 + it * 6 + j] = h[j];
  }
}

// Per-node 32x32 cross attention (fca core). One wave per node.
__global__ void k_fca_attn(const float* __restrict__ q, const float* __restrict__ k,
                           const float* __restrict__ v, float* __restrict__ o, int Nn) {
  const int n = blockIdx.x, j = threadIdx.x;
  const float qj = q[n * 32 + j], kj = k[n * 32 + j], vj = v[n * 32 + j];
  float oi = 0.f;
  for (int i = 0; i < 32; ++i) {
    const float qi = __shfl(qj, i, 32);
    const float s = qi * kj;
    const float m = waveMax(s);
    const float e = expf(s - m);
    const float se = waveSum(e);
    const float num = waveSum(e * vj);
    const float r = num / se;
    if (j == i) oi = r;
  }
  o[n * 32 + j] = oi;
}

__global__ void k_layernorm32(const float* __restrict__ in, const float* __restrict__ g,
                              const float* __restrict__ b, float* __restrict__ out, int Nn) {
  const int n = blockIdx.x, j = threadIdx.x;
  const float x = in[n * 32 + j];
  const float mu = waveSum(x) * (1.f / 32.f);
  const float d = x - mu;
  const float var = waveSum(d * d) * (1.f / 32.f);
  out[n * 32 + j] = d * rsqrtf(var + 1e-5f) * g[j] + b[j];
}

// out[m] = dot(A[m,:], vec)   (one wave per row)
__global__ void k_rowdot(const float* __restrict__ A, int lda, const float* __restrict__ vec,
                         float* __restrict__ out, int K) {
  const int n = blockIdx.x, l = threadIdx.x;
  float s = 0.f;
  for (int k = l; k < K; k += 32) s += A[n * lda + k] * vec[k];
  s = waveSum(s);
  if (l == 0) out[n] = s;
}

// a[e] = leaky_relu(aS[src?src[e]:e] + aD[dst[e]], 0.01)
__global__ void k_edge_logit(const float* __restrict__ aS, const float* __restrict__ aD,
                             const int* __restrict__ srcIdx, const int* __restrict__ dstIdx,
                             float* __restrict__ out, int n) {
  int e = blockIdx.x * blockDim.x + threadIdx.x;
  if (e >= n) return;
  const int s = srcIdx ? srcIdx[e] : e;
  const float v = aS[s] + aD[dstIdx[e]];
  out[e] = v > 0.f ? v : 0.01f * v;
}

__global__ void k_segmax(const float* __restrict__ a, const int* __restrict__ seg,
                         float* __restrict__ m, int n) {
  int i = blockIdx.x * blockDim.x + threadIdx.x;
  if (i < n) atomicMaxF(&m[seg[i]], a[i]);
}
__global__ void k_segexp(const float* __restrict__ a, const int* __restrict__ seg,
                         const float* __restrict__ m, float* __restrict__ e_out,
                         float* __restrict__ ssum, int n) {
  int i = blockIdx.x * blockDim.x + threadIdx.x;
  if (i >= n) return;
  const float e = expf(a[i] - m[seg[i]]);
  e_out[i] = e;
  atomicAdd(&ssum[seg[i]], e);
}
__global__ void k_segnorm(float* __restrict__ alpha, const int* __restrict__ seg,
                          const float* __restrict__ ssum, int n) {
  int i = blockIdx.x * blockDim.x + threadIdx.x;
  if (i < n) alpha[i] = alpha[i] / fmaxf(ssum[seg[i]], 1e-16f);
}

// out[dst[e],:] += alpha[e] * feat[src?src[e]:e, :]
__global__ void k_scatter_wadd(const float* __restrict__ alpha, const int* __restrict__ srcIdx,
                               const int* __restrict__ dstIdx, const float* __restrict__ feat,
                               float* __restrict__ out, int D) {
  const int e = blockIdx.x, d = threadIdx.x;
  const int s = srcIdx ? srcIdx[e] : e;
  atomicAdd(&out[dstIdx[e] * D + d], alpha[e] * feat[s * D + d]);
}

__global__ void k_seg_sum_feat(const float* __restrict__ v, const int* __restrict__ seg,
                               float* __restrict__ out, int D) {
  const int n = blockIdx.x, d = threadIdx.x;
  atomicAdd(&out[seg[n] * D + d], v[n * D + d]);
}

__global__ void k_gru_combine(const float* __restrict__ gi, const float* __restrict__ gh,
                              const float* __restrict__ hprev, float* __restrict__ out,
                              int rows, int D, int doRelu) {
  int idx = blockIdx.x * blockDim.x + threadIdx.x;
  if (idx >= rows * D) return;
  const int n = idx / D, d = idx - n * D;
  const float ir = gi[n * 3 * D + d],          hr = gh[n * 3 * D + d];
  const float iz = gi[n * 3 * D + D + d],      hz = gh[n * 3 * D + D + d];
  const float in_ = gi[n * 3 * D + 2 * D + d], hn = gh[n * 3 * D + 2 * D + d];
  const float r = 1.f / (1.f + expf(-(ir + hr)));
  const float z = 1.f / (1.f + expf(-(iz + hz)));
  const float nn = tanhf(in_ + r * hn);
  float v = (1.f - z) * nn + z * hprev[idx];
  out[idx] = doRelu ? (v > 0.f ? v : 0.f) : v;
}

__global__ void k_bias_act(const float* __restrict__ in, const float* __restrict__ bias,
                           float* __restrict__ out, int rows, int D, int act) {
  int idx = blockIdx.x * blockDim.x + threadIdx.x;
  if (idx >= rows * D) return;
  out[idx] = actf(in[idx] + bias[idx % D], act);
}

__global__ void k_act_inplace(float* __restrict__ p, int n, int act) {
  int i = blockIdx.x * blockDim.x + threadIdx.x;
  if (i < n) p[i] = actf(p[i], act);
}

__global__ void k_mean_div(const float* __restrict__ s, const float* __restrict__ cnt,
                           float* __restrict__ out, int rows, int D) {
  int idx = blockIdx.x * blockDim.x + threadIdx.x;
  if (idx >= rows * D) return;
  out[idx] = s[idx] / fmaxf(cnt[idx / D], 1.f);
}

__global__ void k_copy_cols(const float* __restrict__ src, int lds,
                            float* __restrict__ dst, int ldd, int off, int cols) {
  const int n = blockIdx.x, c = threadIdx.x;
  if (c < cols) dst[n * ldd + off + c] = src[n * lds + c];
}

// ej[e] = concat(xv[src[e]] (160), edge_attr[e] (10))
__global__ void k_gather_concat(const float* __restrict__ xv, const float* __restrict__ ea,
                                const int* __restrict__ src, float* __restrict__ ej) {
  const int e = blockIdx.x, d = threadIdx.x;
  if (d < 160)      ej[e * 170 + d] = xv[src[e] * 160 + d];
  else if (d < 170) ej[e * 170 + d] = ea[e * 10 + (d - 160)];
}

// ---------------------------------------------------------------------------
extern "C" void kernel_launch(void* const* d_in, const int* in_sizes, int n_in,
                              void* d_out, int out_size, void* d_ws, size_t ws_size,
                              hipStream_t stream) {
  const float* x   = (const float*)d_in[0];
  const int*   ei  = (const int*)d_in[1];
  const float* ea  = (const float*)d_in[2];
  const int*   bat = (const int*)d_in[3];
  auto P = [&](int i) { return (const float*)d_in[i]; };

  const int Nn = in_sizes[0] / 86;
  const int E  = in_sizes[1] / 2;
  const int G  = 150;
  const int H  = 160;
  const int* src = ei;
  const int* dst = ei + E;

  // ---- workspace bump allocator -------------------------------------------
  char* wp = (char*)d_ws;
  auto alloc = [&](size_t elems) -> float* {
    float* r = (float*)wp;
    wp += ((elems * sizeof(float) + 255) / 256) * 256;
    return r;
  };

  float* cnt    = alloc(Nn);          // edges per dst node
  float* cntb   = alloc(G);           // nodes per graph
  float* out30  = alloc((size_t)Nn * 30);
  float* x2o    = alloc((size_t)Nn * 32);
  float* msum32 = alloc((size_t)Nn * 32);
  float* rootT32= alloc((size_t)Nn * 32);
  float* x2v    = alloc((size_t)Nn * 32);
  float* x1v    = alloc((size_t)Nn * 32);
  float* fq     = alloc((size_t)Nn * 32);
  float* fk     = alloc((size_t)Nn * 32);
  float* fv     = alloc((size_t)Nn * 32);
  float* fo     = alloc((size_t)Nn * 32);
  float* ft     = alloc((size_t)Nn * 32);
  float* inter  = alloc((size_t)Nn * 32);
  float* gu     = alloc((size_t)Nn * 32);
  float* grp    = alloc((size_t)Nn * 32);
  float* au     = alloc((size_t)Nn * 32);
  float* cat96  = alloc((size_t)Nn * 96);
  float* xx     = alloc((size_t)Nn * 96);
  float* xc0    = alloc((size_t)Nn * H);
  float* xc1    = alloc((size_t)Nn * H);
  float* msum160= alloc((size_t)Nn * H);
  float* rootT160=alloc((size_t)Nn * H);
  float* xc2    = alloc((size_t)Nn * H);
  float* xc3    = alloc((size_t)Nn * H);
  float* xv     = alloc((size_t)Nn * H);
  float* ej     = alloc((size_t)E * 170);
  float* xj     = alloc((size_t)E * H);
  float* al     = alloc(E);
  float* ar     = alloc(Nn);
  float* aedge  = alloc(E);
  float* segm   = alloc(Nn);
  float* segs   = alloc(Nn);
  float* alpha  = alloc(E);
  float* t160   = alloc((size_t)Nn * H);
  float* hmsg   = alloc((size_t)Nn * H);
  float* xin    = alloc((size_t)Nn * H);
  float* gi     = alloc((size_t)Nn * 480);
  float* gh     = alloc((size_t)Nn * 480);
  float* xv2    = alloc((size_t)Nn * H);
  float* hsb    = alloc((size_t)Nn * H);
  float* asb    = alloc(Nn);
  float* adb    = alloc(Nn);
  float* gout   = alloc((size_t)Nn * H);
  float* h2     = alloc((size_t)Nn * H);
  float* xv3    = alloc((size_t)Nn * H);
  float* outg   = alloc((size_t)G * H);
  float* hsM    = alloc((size_t)Nn * H);
  float* asM    = alloc(Nn);
  float* hdM    = alloc((size_t)G * H);
  float* adM    = alloc(G);
  float* aM     = alloc(Nn);
  float* mG     = alloc(G);
  float* sG     = alloc(G);
  float* alphaM = alloc(Nn);
  float* gsum   = alloc((size_t)G * H);
  float* hG     = alloc((size_t)G * H);
  float* giG    = alloc((size_t)G * 480);
  float* ghG    = alloc((size_t)G * 480);
  float* xg     = alloc((size_t)G * 320);
  float* isum   = alloc((size_t)G * 32);
  float* imean  = alloc((size_t)G * 32);
  float* gpool  = alloc((size_t)G * 128);
  float* ycat   = alloc((size_t)G * 448);
  float* f1     = alloc((size_t)G * 512);
  float* f2     = alloc((size_t)G * 256);
  float* f3     = alloc((size_t)G * 64);
  (void)ws_size; (void)n_in; (void)out_size;

  // ---- helpers -------------------------------------------------------------
  auto fill = [&](float* p, float v, int n) {
    k_fill<<<(n + 255) / 256, 256, 0, stream>>>(p, v, n);
  };
  auto gemm = [&](const float* A, int lda, const float* B, int ldb,
                  const float* bias, const float* add, int ldadd,
                  float* C, int ldc, int M, int K, int Nc, int act) {
    dim3 g((unsigned)((Nc + 15) / 16), (unsigned)((M + 15) / 16));
    k_gemm<<<g, 32, 0, stream>>>(A, lda, B, ldb, bias, add, ldadd, C, ldc, M, K, Nc, act);
  };
  auto segsoftmax = [&](const float* logits, const int* seg, int items, int nseg,
                        float* mbuf, float* sbuf, float* aOut) {
    fill(mbuf, -1e30f, nseg);
    k_segmax<<<(items + 255) / 256, 256, 0, stream>>>(logits, seg, mbuf, items);
    fill(sbuf, 0.f, nseg);
    k_segexp<<<(items + 255) / 256, 256, 0, stream>>>(logits, seg, mbuf, aOut, sbuf, items);
    k_segnorm<<<(items + 255) / 256, 256, 0, stream>>>(aOut, seg, sbuf, items);
  };
  auto fca = [&](const float* qin, int ldq, int dq, const float* kvin, int ldkv, int dkv,
                 int pi, float* outb) {
    gemm(qin, ldq, P(pi + 0), 32, P(pi + 1), nullptr, 0, fq, 32, Nn, dq, 32, ACT_NONE);
    gemm(kvin, ldkv, P(pi + 2), 32, P(pi + 3), nullptr, 0, fk, 32, Nn, dkv, 32, ACT_NONE);
    gemm(kvin, ldkv, P(pi + 4), 32, P(pi + 5), nullptr, 0, fv, 32, Nn, dkv, 32, ACT_NONE);
    k_fca_attn<<<Nn, 32, 0, stream>>>(fq, fk, fv, fo, Nn);
    gemm(fo, 32, P(pi + 6), 32, P(pi + 7), fq, 32, ft, 32, Nn, 32, 32, ACT_NONE);
    k_layernorm32<<<Nn, 32, 0, stream>>>(ft, P(pi + 8), P(pi + 9), outb, Nn);
  };
  auto nnconv = [&](const float* xin_, int ldx, int cin, int cout, int pi,
                    float* msum, float* rootT, float* outb) {
    fill(msum, 0.f, Nn * cout);
    // one wave per 16-wide cout tile, 16 edges per block
    k_nnconv_wmma<<<(E + 15) / 16, 32 * (cout / 16), 0, stream>>>(
        xin_, ldx, src, dst, ea, P(pi + 0), P(pi + 1), P(pi + 2), P(pi + 3),
        msum, cin, cout, E);
    gemm(xin_, ldx, P(pi + 4), cout, nullptr, nullptr, 0, rootT, cout, Nn, cin, cout, ACT_NONE);
    k_nnconv_fin<<<(Nn * cout + 255) / 256, 256, 0, stream>>>(msum, cnt, rootT, P(pi + 5),
                                                              outb, Nn, cout);
  };
  auto gru = [&](const float* xin_, const float* h, int piw, float* giB, float* ghB,
                 float* outb, int rows, int doRelu) {
    gemm(xin_, H, P(piw + 0), 480, P(piw + 1), nullptr, 0, giB, 480, rows, H, 480, ACT_NONE);
    gemm(h, H, P(piw + 2), 480, P(piw + 3), nullptr, 0, ghB, 480, rows, H, 480, ACT_NONE);
    k_gru_combine<<<(rows * H + 255) / 256, 256, 0, stream>>>(giB, ghB, h, outb, rows, H, doRelu);
  };

  // ---- counts --------------------------------------------------------------
  fill(cnt, 0.f, Nn);
  k_count_seg<<<(E + 255) / 256, 256, 0, stream>>>(dst, cnt, E);
  fill(cntb, 0.f, G);
  k_count_seg<<<(Nn + 255) / 256, 256, 0, stream>>>(bat, cntb, Nn);

  // ---- CfC + x22 -----------------------------------------------------------
  k_cfc<<<(Nn + 127) / 128, 128, 0, stream>>>(x + 42, 86, P(4), P(5), P(6), P(7),
                                              P(8), P(9), P(10), P(11), out30, Nn);
  gemm(out30, 30, P(12), 32, P(13), nullptr, 0, x2o, 32, Nn, 30, 32, ACT_RELU);

  // ---- NNConvs a21 / a11 ---------------------------------------------------
  nnconv(x2o, 32, 32, 32, 14, msum32, rootT32, x2v);   // a21
  nnconv(x,   86, 41, 32, 20, msum32, rootT32, x1v);   // a11 (x1 = x[:, :41])

  // ---- FCA chain -----------------------------------------------------------
  fca(x + 67, 86, 19, x + 67, 86, 19, 26, inter);      // inter = fca(g, g)
  fca(x + 48, 86, 19, inter, 32, 32, 36, gu);          // gu    = fca(x3, inter)
  fca(x1v, 32, 32, gu, 32, 32, 46, grp);               // grp   = fca(x1v, gu)
  fca(x2v, 32, 32, grp, 32, 32, 56, au);               // au    = fca(x2v, grp)

  // ---- xx / xc -------------------------------------------------------------
  k_copy_cols<<<Nn, 32, 0, stream>>>(gu, 32, cat96, 96, 0, 32);
  k_copy_cols<<<Nn, 32, 0, stream>>>(grp, 32, cat96, 96, 32, 32);
  k_copy_cols<<<Nn, 32, 0, stream>>>(au, 32, cat96, 96, 64, 32);
  gemm(cat96, 96, P(66), 96, P(67), nullptr, 0, xx, 96, Nn, 96, 96, ACT_RELU);

  k_copy_cols<<<Nn, 32, 0, stream>>>(x1v, 32, xc0, H, 0, 32);
  k_copy_cols<<<Nn, 32, 0, stream>>>(x2v, 32, xc0, H, 32, 32);
  k_copy_cols<<<Nn, 96, 0, stream>>>(xx, 96, xc0, H, 64, 96);
  gemm(xc0, H, P(68), H, P(69), nullptr, 0, xc1, H, Nn, H, H, ACT_RELU);

  nnconv(xc1, H, H, H, 70, msum160, rootT160, xc2);    // sub1
  nnconv(xc2, H, H, H, 76, msum160, rootT160, xc3);    // sub2

  // ---- AttentiveFP ---------------------------------------------------------
  gemm(xc3, H, P(82), H, P(83), nullptr, 0, xv, H, Nn, H, H, ACT_LEAKY);   // lin1
  k_gather_concat<<<E, 192, 0, stream>>>(xv, ea, src, ej);
  gemm(ej, 170, P(84), H, nullptr, nullptr, 0, xj, H, E, 170, H, ACT_LEAKY); // gate_lin1
  k_rowdot<<<E, 32, 0, stream>>>(xj, H, P(85), al, H);                     // gate_att_l
  k_rowdot<<<Nn, 32, 0, stream>>>(xv, H, P(86), ar, H);                    // gate_att_r
  k_edge_logit<<<(E + 255) / 256, 256, 0, stream>>>(al, ar, nullptr, dst, aedge, E);
  segsoftmax(aedge, dst, E, Nn, segm, segs, alpha);
  gemm(xv, H, P(87), H, nullptr, nullptr, 0, t160, H, Nn, H, H, ACT_NONE); // gate_lin2
  fill(hmsg, 0.f, Nn * H);
  k_scatter_wadd<<<E, H, 0, stream>>>(alpha, src, dst, t160, hmsg, H);
  k_bias_act<<<(Nn * H + 255) / 256, 256, 0, stream>>>(hmsg, P(88), xin, Nn, H, ACT_ELU);
  gru(xin, xv, 89, gi, gh, xv2, Nn, 1);                                    // gru1 (+relu)

  // gat conv1
  gemm(xv2, H, P(93), H, nullptr, nullptr, 0, hsb, H, Nn, H, H, ACT_NONE);
  k_rowdot<<<Nn, 32, 0, stream>>>(hsb, H, P(94), asb, H);
  k_rowdot<<<Nn, 32, 0, stream>>>(hsb, H, P(95), adb, H);
  k_edge_logit<<<(E + 255) / 256, 256, 0, stream>>>(asb, adb, src, dst, aedge, E);
  segsoftmax(aedge, dst, E, Nn, segm, segs, alpha);
  fill(gout, 0.f, Nn * H);
  k_scatter_wadd<<<E, H, 0, stream>>>(alpha, src, dst, hsb, gout, H);
  k_bias_act<<<(Nn * H + 255) / 256, 256, 0, stream>>>(gout, P(96), h2, Nn, H, ACT_ELU);
  gru(h2, xv2, 97, gi, gh, xv3, Nn, 1);                                    // gru2 (+relu)

  // graph pooling + mol loop
  fill(outg, 0.f, G * H);
  k_seg_sum_feat<<<Nn, H, 0, stream>>>(xv3, bat, outg, H);
  k_act_inplace<<<(G * H + 255) / 256, 256, 0, stream>>>(outg, G * H, ACT_RELU);

  gemm(xv3, H, P(101), H, nullptr, nullptr, 0, hsM, H, Nn, H, H, ACT_NONE); // mol_conv.w
  k_rowdot<<<Nn, 32, 0, stream>>>(hsM, H, P(102), asM, H);                  // att_src
  for (int it = 0; it < 2; ++it) {
    gemm(outg, H, P(101), H, nullptr, nullptr, 0, hdM, H, G, H, H, ACT_NONE);
    k_rowdot<<<G, 32, 0, stream>>>(hdM, H, P(103), adM, H);                 // att_dst
    k_edge_logit<<<(Nn + 255) / 256, 256, 0, stream>>>(asM, adM, nullptr, bat, aM, Nn);
    segsoftmax(aM, bat, Nn, G, mG, sG, alphaM);
    fill(gsum, 0.f, G * H);
    k_scatter_wadd<<<Nn, H, 0, stream>>>(alphaM, nullptr, bat, hsM, gsum, H);
    k_bias_act<<<(G * H + 255) / 256, 256, 0, stream>>>(gsum, P(104), hG, G, H, ACT_ELU);
    gru(hG, outg, 105, giG, ghG, outg, G, 1);                               // mol_gru (+relu)
  }
  gemm(outg, H, P(109), 320, P(110), nullptr, 0, xg, 320, G, H, 320, ACT_NONE); // lin2

  // ---- group pool + head ---------------------------------------------------
  fill(isum, 0.f, G * 32);
  k_seg_sum_feat<<<Nn, 32, 0, stream>>>(inter, bat, isum, 32);
  k_mean_div<<<(G * 32 + 255) / 256, 256, 0, stream>>>(isum, cntb, imean, G, 32);
  gemm(imean, 32, P(111), 128, P(112), nullptr, 0, gpool, 128, G, 32, 128, ACT_RELU);

  k_copy_cols<<<G, 320, 0, stream>>>(xg, 320, ycat, 448, 0, 320);
  k_copy_cols<<<G, 128, 0, stream>>>(gpool, 128, ycat, 448, 320, 128);
  gemm(ycat, 448, P(113), 512, P(114), nullptr, 0, f1, 512, G, 448, 512, ACT_RELU);
  gemm(f1, 512, P(115), 256, P(116), nullptr, 0, f2, 256, G, 512, 256, ACT_RELU);
  gemm(f2, 256, P(117), 64, P(118), nullptr, 0, f3, 64, G, 256, 64, ACT_RELU);
  gemm(f3, 64, P(119), 1, P(120), nullptr, 0, (float*)d_out, 1, G, 64, 1, ACT_NONE);
}